// TsodycsMarkramSynapse_18313740550501
// MI455X (gfx1250) — compile-verified
//
#include <hip/hip_runtime.h>
#include <stdint.h>

// Problem constants (fixed by the reference)
#define B_     4
#define T_     1000
#define N_     8192
#define BN_    (B_ * N_)        // 32768 synapses
#define BLOCK_ 256              // 8 waves of 32
#define NBLK_  (BN_ / BLOCK_)   // 128 workgroups
#define DEPTH_ 32               // async prefetch ring depth (ASYNCcnt is 6 bits; 32 < 64)
#define GRP_   8                // timesteps per __syncthreads flush
#define DT_    0.1f

// ---------------------------------------------------------------------------
// Phase 1: per-synapse sequential scan with CDNA5 async global->LDS staging.
// Each thread owns one (b, n) column; FR[t] for that column is prefetched
// DEPTH_ steps ahead with global_load_async_to_lds_b32 into a ring buffer.
// Per timestep: wave32 shfl-xor reduction of (g, g*E), lane0 banks the pair
// in LDS; every GRP_ steps one barrier + 8 threads flush block partials.
// ---------------------------------------------------------------------------
__global__ __launch_bounds__(BLOCK_) void tm_scan_kernel(
    const float* __restrict__ FR,    const float* __restrict__ pconn,
    const float* __restrict__ Erev,  const float* __restrict__ gmax,
    const float* __restrict__ tauf,  const float* __restrict__ taud,
    const float* __restrict__ taur,  const float* __restrict__ Uinc,
    float* __restrict__ partials)
{
  __shared__ float frbuf[DEPTH_ * BLOCK_];            // 32 KB async staging ring
  __shared__ float wred[2][GRP_][2 * (BLOCK_ / 32)];  // double-banked wave partials

  const int tid  = threadIdx.x;
  const int blk  = blockIdx.x;
  const int e    = blk * BLOCK_ + tid;   // [0, 32768)
  const int b    = e >> 13;              // e / N_
  const int n    = e & (N_ - 1);
  const int wave = tid >> 5;
  const int lane = tid & 31;

  // Per-unit constants (one coalesced load each, then registers forever).
  const float td    = taud[n];
  const float tr    = taur[n];
  const float e_d   = expf(-DT_ / td);
  const float e_r   = expf(-DT_ / tr);
  const float e_f   = expf(-DT_ / tauf[n]);
  const float tau1r = (td != tr) ? (td / (td - tr)) : 1e-13f;
  const float pc    = pconn[n];
  const float ui    = Uinc[n];
  const float cg    = fmaxf(gmax[n], 0.0f);   // relu(gsyn_max)
  const float cge   = cg * Erev[n];

  const float* base = FR + (size_t)b * T_ * N_ + n;   // stride N_ floats per t

  // Prologue: fill the async ring DEPTH_ steps deep (ASYNCcnt = DEPTH_).
  for (int d = 0; d < DEPTH_; ++d) {
    uint32_t loff = (uint32_t)(uintptr_t)&frbuf[d * BLOCK_ + tid];
    uint64_t g    = (uint64_t)(uintptr_t)(base + (size_t)d * N_);
    asm volatile("global_load_async_to_lds_b32 %0, %1, off"
                 :: "v"(loff), "v"(g) : "memory");
  }

  float R = 1.0f, U = 0.0f, X = 0.0f;

  for (int t = 0; t < T_; ++t) {
    // Async loads complete in order: count <= DEPTH_-1 => step-t data landed.
    asm volatile("s_wait_asynccnt 31" ::: "memory");
    const int slot = t & (DEPTH_ - 1);
    const float fr = frbuf[slot * BLOCK_ + tid];

    // Tsodyks-Markram update (fp32, matches reference expression order).
    const float frn = fr * pc;
    const float y_  = R * e_d;
    const float x_  = 1.0f + (X - 1.0f + tau1r * U) * e_r - tau1r * U;
    const float u_  = U * e_f;
    const float Un  = u_ + ui * (1.0f - u_) * frn;
    const float a   = Un * x_ * frn;
    R = y_ + a;
    X = x_ - a;
    U = Un;

    // Refill the just-consumed slot for step t+DEPTH_ (clamped near the end
    // so the outstanding-count pattern stays uniform).
    {
      int tp = t + DEPTH_;
      if (tp > T_ - 1) tp = T_ - 1;
      uint32_t loff = (uint32_t)(uintptr_t)&frbuf[slot * BLOCK_ + tid];
      uint64_t g    = (uint64_t)(uintptr_t)(base + (size_t)tp * N_);
      asm volatile("global_load_async_to_lds_b32 %0, %1, off"
                   :: "v"(loff), "v"(g) : "memory");
    }

    // Per-t contribution and wave32 reduction (all lanes active).
    float g1 = cg  * X;   // gsyn
    float g2 = cge * X;   // gsyn * Erev
    #pragma unroll
    for (int off = 16; off > 0; off >>= 1) {
      g1 += __shfl_xor(g1, off, 32);
      g2 += __shfl_xor(g2, off, 32);
    }
    const int bank  = (t >> 3) & 1;
    const int wslot = t & (GRP_ - 1);
    if (lane == 0) {
      wred[bank][wslot][2 * wave + 0] = g1;
      wred[bank][wslot][2 * wave + 1] = g2;
    }

    // One barrier per GRP_ steps; 8 threads flush 8 timesteps of partials.
    if (wslot == GRP_ - 1) {          // uniform branch (depends on t only)
      __syncthreads();
      if (tid < GRP_) {
        const int tt = t - (GRP_ - 1) + tid;
        float sg = 0.0f, sge = 0.0f;
        #pragma unroll
        for (int w = 0; w < BLOCK_ / 32; ++w) {
          sg  += wred[bank][tid][2 * w + 0];
          sge += wred[bank][tid][2 * w + 1];
        }
        const size_t o = ((size_t)tt * NBLK_ + blk) * 2;
        partials[o + 0] = sg;
        partials[o + 1] = sge;
      }
    }
  }
}

// ---------------------------------------------------------------------------
// Phase 2: reduce the 128 block-partials per timestep (fixed tree order,
// fully deterministic) and emit E = sum(gE)/sum(g), tau = Cm/sum(g).
// ---------------------------------------------------------------------------
__global__ __launch_bounds__(NBLK_) void tm_finish_kernel(
    const float* __restrict__ partials, const float* __restrict__ Cm,
    float* __restrict__ out)
{
  const int t   = blockIdx.x;
  const int tid = threadIdx.x;           // 128 threads = 4 waves
  __shared__ float wr[2 * (NBLK_ / 32)];

  const float2 p = ((const float2*)partials)[(size_t)t * NBLK_ + tid];
  float sg = p.x, sge = p.y;
  #pragma unroll
  for (int off = 16; off > 0; off >>= 1) {
    sg  += __shfl_xor(sg,  off, 32);
    sge += __shfl_xor(sge, off, 32);
  }
  if ((tid & 31) == 0) {
    wr[2 * (tid >> 5) + 0] = sg;
    wr[2 * (tid >> 5) + 1] = sge;
  }
  __syncthreads();
  if (tid == 0) {
    float g = 0.0f, ge = 0.0f;
    #pragma unroll
    for (int w = 0; w < NBLK_ / 32; ++w) {
      g  += wr[2 * w + 0];
      ge += wr[2 * w + 1];
    }
    out[2 * t + 0] = ge / g;        // E
    out[2 * t + 1] = Cm[0] / g;     // tau
  }
}

extern "C" void kernel_launch(void* const* d_in, const int* in_sizes, int n_in,
                              void* d_out, int out_size, void* d_ws, size_t ws_size,
                              hipStream_t stream) {
  (void)in_sizes; (void)n_in; (void)out_size; (void)ws_size;
  const float* FR    = (const float*)d_in[0];
  const float* pconn = (const float*)d_in[1];
  const float* Erev  = (const float*)d_in[2];
  const float* gmax  = (const float*)d_in[3];
  const float* tauf  = (const float*)d_in[4];
  const float* taud  = (const float*)d_in[5];
  const float* taur  = (const float*)d_in[6];
  const float* Uinc  = (const float*)d_in[7];
  const float* Cm    = (const float*)d_in[8];
  float* partials = (float*)d_ws;   // needs T_*NBLK_*2 floats = 1,024,000 B

  tm_scan_kernel<<<NBLK_, BLOCK_, 0, stream>>>(FR, pconn, Erev, gmax,
                                               tauf, taud, taur, Uinc, partials);
  tm_finish_kernel<<<T_, NBLK_, 0, stream>>>(partials, Cm, (float*)d_out);
}